// Loss_Q_62259845922881
// MI455X (gfx1250) — compile-verified
//
#include <hip/hip_runtime.h>

typedef __attribute__((ext_vector_type(2))) float v2f;
typedef __attribute__((ext_vector_type(8))) float v8f;
typedef __attribute__((ext_vector_type(4))) unsigned int v4u;
typedef __attribute__((ext_vector_type(8))) int v8i;
typedef __attribute__((ext_vector_type(4))) int v4i;

#define NBAND   8
#define KS      41
#define NCH     14      // K-chunks of 4 covering the 56-column x-window
#define PSTR    56      // LDS row stride in floats (TDM writes unpadded rows)
#define PROWS   72      // 32 output rows + 40 halo
#define IMG_IN  552
#define IMG_MID 512
#define IMG_OUT 481

// ---------------------------------------------------------------------------
// Kernel 0: materialize Toeplitz-band B operands in the exact WMMA B-matrix
// (4x16 f32, 2 VGPR) lane layout:
//   lane l: n = l&15, j0 = 4c + (l>=16 ? 2 : 0)
//   elem0 = w[ky][j0-n]   (0 outside band),  elem1 = w[ky][j0+1-n]
// Bmat[b][ky][c][lane][2]  -> one coalesced global_load_b64 per B chunk.
// ---------------------------------------------------------------------------
__global__ __launch_bounds__(32)
void build_toeplitz(const float* __restrict__ mtf, float* __restrict__ Bmat) {
    int id   = blockIdx.x;                 // b*KS*NCH + ky*NCH + c
    int c    = id % NCH;
    int ky   = (id / NCH) % KS;
    int b    = id / (NCH * KS);
    int lane = threadIdx.x;
    int n    = lane & 15;
    int j0   = 4 * c + ((lane >= 16) ? 2 : 0);
    const float* wrow = mtf + ((size_t)b * KS + ky) * KS;
    int d0 = j0 - n, d1 = j0 + 1 - n;
    float v0 = (d0 >= 0 && d0 < KS) ? wrow[d0] : 0.0f;
    float v1 = (d1 >= 0 && d1 < KS) ? wrow[d1] : 0.0f;
    float* dst = Bmat + (size_t)id * 64 + lane * 2;
    dst[0] = v0;
    dst[1] = v1;
}

// ---------------------------------------------------------------------------
// Kernel 1: 41x41 depthwise MTF conv via V_WMMA_F32_16X16X4_F32.
// One wave computes a 32x16 output block (two 16x16 y-tiles) so each Toeplitz
// B chunk load feeds 2 WMMAs; 4 accumulator chains hide WMMA latency.
// 72x56 input patch staged into LDS by the Tensor Data Mover (TENSORcnt).
// ---------------------------------------------------------------------------
__global__ __launch_bounds__(32)
void mtf_conv_wmma(const float* __restrict__ x, const float* __restrict__ Bmat,
                   float* __restrict__ obuf) {
    __shared__ __align__(16) float patch[PROWS * PSTR];

    const int lane = threadIdx.x;
    const int x0   = blockIdx.x * 16;
    const int y0   = blockIdx.y * 32;
    const int nb   = blockIdx.z;            // n*8 + band
    const int band = nb & (NBAND - 1);

    const float* src = x + (size_t)nb * IMG_IN * IMG_IN + (size_t)y0 * IMG_IN + x0;

#if __has_builtin(__builtin_amdgcn_tensor_load_to_lds)
    {
        // Tensor DMA descriptor (ISA 8.3/8.4): 72-row x 56-col f32 tile from a
        // 552-stride tensor into contiguous LDS rows.
        unsigned long long ga = (unsigned long long)(const void*)src;
        unsigned lds_base = (unsigned)(unsigned long long)(void*)patch;
        v4u g0 = { 1u,                                   // count=1, user mode
                   lds_base,                             // lds_addr
                   (unsigned)ga,                         // global_addr[31:0]
                   (unsigned)((ga >> 32) & 0x1FFFFFFu) | (2u << 30) }; // addr[56:32] | type=2
        v8i g1 = { (int)(2u << 16),                      // data_size = 4B
                   (int)(552u << 16),                    // tensor_dim0 = 552 (lo16 in hi half)
                   (int)(552u << 16),                    // tensor_dim0 hi=0 | tensor_dim1 lo16
                   (int)(56u << 16),                     // tensor_dim1 hi=0 | tile_dim0 = 56
                   (int)PROWS,                           // tile_dim1 = 72, tile_dim2 = 0
                   (int)552,                             // tensor_dim0_stride = 552
                   0, 0 };
        v4i gz4 = { 0, 0, 0, 0 };                        // group2 (2-D tile: unused)
        v8i gz8 = { 0, 0, 0, 0, 0, 0, 0, 0 };            // group3/extra (unused)
        __builtin_amdgcn_tensor_load_to_lds(g0, g1, gz4, gz4, gz8, 0);
        __builtin_amdgcn_s_wait_tensorcnt(0);
    }
#else
    for (int i = lane; i < PROWS * 14; i += 32) {
        int r = i / 14, c4 = i % 14;
        float4 v = *(const float4*)(src + (size_t)r * IMG_IN + c4 * 4);
        *(float4*)&patch[r * PSTR + c4 * 4] = v;
    }
#endif
    __syncthreads();

    const int m    = lane & 15;             // A-matrix M per lane
    const int moff = (lane >= 16) ? 2 : 0;  // K offset of VGPR0 in upper half
    const float* Arow = &patch[m * PSTR + moff];
    const float* Bb   = Bmat + (size_t)band * KS * NCH * 64 + lane * 2;

    v8f acc0 = {0.f,0.f,0.f,0.f,0.f,0.f,0.f,0.f};
    v8f acc1 = {0.f,0.f,0.f,0.f,0.f,0.f,0.f,0.f};
    v8f acc2 = {0.f,0.f,0.f,0.f,0.f,0.f,0.f,0.f};
    v8f acc3 = {0.f,0.f,0.f,0.f,0.f,0.f,0.f,0.f};

    for (int ky = 0; ky < KS - 1; ky += 2) {
        const float* p00 = Arow + ky * PSTR;            // tile0, ky
        const float* p01 = p00 + PSTR;                  // tile0, ky+1
        const float* p10 = p00 + 16 * PSTR;             // tile1, ky
        const float* p11 = p10 + PSTR;                  // tile1, ky+1
        const float* b0r = Bb + (size_t)ky * NCH * 64;
        const float* b1r = b0r + (size_t)NCH * 64;
        for (int c = 0; c < NCH; ++c) {
            v2f w0 = *(const v2f*)(b0r + (size_t)c * 64);   // shared by 2 WMMAs
            v2f a00 = *(const v2f*)(p00 + 4 * c);
            acc0 = __builtin_amdgcn_wmma_f32_16x16x4_f32(
                false, a00, false, w0, (short)0, acc0, false, false);
            v2f a10 = *(const v2f*)(p10 + 4 * c);
            acc2 = __builtin_amdgcn_wmma_f32_16x16x4_f32(
                false, a10, false, w0, (short)0, acc2, false, false);
            v2f w1 = *(const v2f*)(b1r + (size_t)c * 64);
            v2f a01 = *(const v2f*)(p01 + 4 * c);
            acc1 = __builtin_amdgcn_wmma_f32_16x16x4_f32(
                false, a01, false, w1, (short)0, acc1, false, false);
            v2f a11 = *(const v2f*)(p11 + 4 * c);
            acc3 = __builtin_amdgcn_wmma_f32_16x16x4_f32(
                false, a11, false, w1, (short)0, acc3, false, false);
        }
    }
    {   // ky = 40 (odd tail)
        const float* p0 = Arow + (KS - 1) * PSTR;
        const float* p1 = p0 + 16 * PSTR;
        const float* br = Bb + (size_t)(KS - 1) * NCH * 64;
        for (int c = 0; c < NCH; ++c) {
            v2f w  = *(const v2f*)(br + (size_t)c * 64);
            v2f a0 = *(const v2f*)(p0 + 4 * c);
            acc0 = __builtin_amdgcn_wmma_f32_16x16x4_f32(
                false, a0, false, w, (short)0, acc0, false, false);
            v2f a1 = *(const v2f*)(p1 + 4 * c);
            acc2 = __builtin_amdgcn_wmma_f32_16x16x4_f32(
                false, a1, false, w, (short)0, acc2, false, false);
        }
    }
    v8f d0 = acc0 + acc1;
    v8f d1 = acc2 + acc3;

    // C/D layout: VGPR r -> M = r (lanes 0-15) / 8+r (lanes 16-31), N = lane&15
    float* dst = obuf + (size_t)nb * IMG_MID * IMG_MID + (size_t)y0 * IMG_MID + x0;
    const int nn    = lane & 15;
    const int mbase = (lane >= 16) ? 8 : 0;
    for (int r = 0; r < 8; ++r)
        dst[(size_t)(mbase + r) * IMG_MID + nn] = d0[r];
    float* dst2 = dst + (size_t)16 * IMG_MID;
    for (int r = 0; r < 8; ++r)
        dst2[(size_t)(mbase + r) * IMG_MID + nn] = d1[r];
}

// ---------------------------------------------------------------------------
// Kernel 2: fused 32x32 box sums (sliding window, O(1)/row vertically) +
// quality map + deterministic block reduction. One block per (image, band).
// ---------------------------------------------------------------------------
__global__ __launch_bounds__(512)
void boxsum_quality(const float* __restrict__ obuf, const float* __restrict__ labels,
                    float* __restrict__ partials) {
    __shared__ float sh[5 * 512];
    const int nb = blockIdx.x;
    const int x  = threadIdx.x;
    const float* po = obuf   + (size_t)nb * IMG_MID * IMG_MID;
    const float* pl = labels + (size_t)nb * IMG_MID * IMG_MID;

    // per-column running sums over the 32-row window
    float c0 = 0.f, c1 = 0.f, c2 = 0.f, c3 = 0.f, c4 = 0.f;
    for (int y = 0; y < 32; ++y) {
        float o = po[(size_t)y * IMG_MID + x];
        float l = pl[(size_t)y * IMG_MID + x];
        c0 += o; c1 += l; c2 += o * o; c3 += l * l; c4 += o * l;
    }

    float qacc = 0.0f;
    for (int yo = 0; yo < IMG_OUT; ++yo) {
        sh[x]        = c0; sh[512 + x]  = c1; sh[1024 + x] = c2;
        sh[1536 + x] = c3; sh[2048 + x] = c4;
        __syncthreads();
        if (x < IMG_OUT) {
            float s0 = 0.f, s1 = 0.f, s2 = 0.f, s3 = 0.f, s4 = 0.f;
            for (int j = 0; j < 32; ++j) {
                s0 += sh[x + j];        s1 += sh[512 + x + j];
                s2 += sh[1024 + x + j]; s3 += sh[1536 + x + j];
                s4 += sh[2048 + x + j];
            }
            const float Nn  = 1024.0f;                 // BS*BS
            float mul = s0 * s1;
            float sq  = s0 * s0 + s1 * s1;
            float num = 4.0f * (Nn * s4 - mul) * mul;
            float dt  = Nn * (s2 + s3) - sq;
            float den = dt * sq;
            float q = 1.0f;
            if (dt == 0.0f && sq != 0.0f) q = 2.0f * mul / sq;
            if (den != 0.0f)              q = num / den;
            qacc += q;
        }
        __syncthreads();
        if (yo < IMG_OUT - 1) {
            float on = po[(size_t)(yo + 32) * IMG_MID + x];
            float ln = pl[(size_t)(yo + 32) * IMG_MID + x];
            float oo = po[(size_t)yo * IMG_MID + x];
            float lo = pl[(size_t)yo * IMG_MID + x];
            c0 += on - oo;           c1 += ln - lo;
            c2 += on * on - oo * oo; c3 += ln * ln - lo * lo;
            c4 += on * ln - oo * lo;
            if (yo + 34 < IMG_MID) {      // global_prefetch_b8 for next rows
                __builtin_prefetch(po + (size_t)(yo + 34) * IMG_MID + x, 0, 1);
                __builtin_prefetch(pl + (size_t)(yo + 34) * IMG_MID + x, 0, 1);
            }
        }
    }

    // deterministic tree reduction
    sh[x] = qacc;
    __syncthreads();
    for (int s = 256; s > 0; s >>= 1) {
        if (x < s) sh[x] += sh[x + s];
        __syncthreads();
    }
    if (x == 0) partials[nb] = sh[0];
}

// ---------------------------------------------------------------------------
// Kernel 3: sequential (deterministic) finalize:  loss = 1 - mean(quality)
// ---------------------------------------------------------------------------
__global__ void finalize_loss(const float* __restrict__ partials,
                              float* __restrict__ out) {
    if (threadIdx.x == 0 && blockIdx.x == 0) {
        float s = 0.0f;
        for (int i = 0; i < 32; ++i) s += partials[i];
        out[0] = 1.0f - s / 7403552.0f;   // 4 * 8 * 481 * 481
    }
}

extern "C" void kernel_launch(void* const* d_in, const int* in_sizes, int n_in,
                              void* d_out, int out_size, void* d_ws, size_t ws_size,
                              hipStream_t stream) {
    const float* outputs = (const float*)d_in[0];  // [4,8,552,552]
    const float* labels  = (const float*)d_in[1];  // [4,8,512,512]
    const float* mtf     = (const float*)d_in[2];  // [8,1,41,41]

    float* obuf  = (float*)d_ws;                          // 33.55 MB
    float* Bmat  = obuf + (size_t)32 * 512 * 512;         // 1.18 MB
    float* parts = Bmat + (size_t)NBAND * KS * NCH * 64;  // 128 B

    build_toeplitz<<<NBAND * KS * NCH, 32, 0, stream>>>(mtf, Bmat);
    mtf_conv_wmma<<<dim3(32, 16, 32), 32, 0, stream>>>(outputs, Bmat, obuf);
    boxsum_quality<<<32, 512, 0, stream>>>(obuf, labels, parts);
    finalize_loss<<<1, 32, 0, stream>>>(parts, (float*)d_out);
}